// LevelSetLayer1D_11536282157684
// MI455X (gfx1250) — compile-verified
//
#include <hip/hip_runtime.h>
#include <hip/hip_bf16.h>

// Problem: 0-dim sublevel persistence on a path graph (P vertices).
// Equivalent closed form (prominence pairing) -> fully parallel.

#define P_N   2097152          // number of vertices
#define BSZ   256              // elements per block (level-0 granule)
#define NB    8192             // P_N / BSZ
#define LEVELS 14              // sparse-table levels j = 0..13 (2^13 = NB)

// Monotone 64-bit key reproducing stable argsort order: (f asc, index asc).
__device__ __forceinline__ unsigned long long kkey(float v, unsigned int idx) {
    unsigned int u = __float_as_uint(v);
    unsigned int s = (u & 0x80000000u) ? ~u : (u | 0x80000000u);
    return ((unsigned long long)s << 32) | (unsigned long long)idx;
}

// ---------------- Kernel A: per-block min/max key (sparse-table level 0) ----
__global__ void block_minmax_kernel(const float* __restrict__ f,
                                    unsigned long long* __restrict__ minTab,
                                    unsigned long long* __restrict__ maxTab) {
    __shared__ unsigned long long smin[BSZ];
    __shared__ unsigned long long smax[BSZ];
    int i = blockIdx.x * BSZ + threadIdx.x;
    unsigned long long k = kkey(f[i], (unsigned int)i);
    smin[threadIdx.x] = k;
    smax[threadIdx.x] = k;
    __syncthreads();
    for (int off = BSZ / 2; off > 0; off >>= 1) {
        if ((int)threadIdx.x < off) {
            unsigned long long a = smin[threadIdx.x + off];
            if (a < smin[threadIdx.x]) smin[threadIdx.x] = a;
            unsigned long long b = smax[threadIdx.x + off];
            if (b > smax[threadIdx.x]) smax[threadIdx.x] = b;
        }
        __syncthreads();
    }
    if (threadIdx.x == 0) {
        minTab[blockIdx.x] = smin[0];
        maxTab[blockIdx.x] = smax[0];
    }
}

// ---------------- Kernel B: build block-level sparse tables (1 workgroup) ---
__global__ void build_sparse_kernel(unsigned long long* __restrict__ minTab,
                                    unsigned long long* __restrict__ maxTab) {
    for (int lev = 1; lev < LEVELS; ++lev) {
        int half = 1 << (lev - 1);
        __syncthreads();  // previous level fully written
        for (int x = threadIdx.x; x < NB; x += blockDim.x) {
            int x2 = x + half; if (x2 > NB - 1) x2 = NB - 1;  // clamp: unread entries
            unsigned long long a = minTab[(lev - 1) * NB + x];
            unsigned long long b = minTab[(lev - 1) * NB + x2];
            minTab[lev * NB + x] = (a < b) ? a : b;
            unsigned long long c = maxTab[(lev - 1) * NB + x];
            unsigned long long d = maxTab[(lev - 1) * NB + x2];
            maxTab[lev * NB + x] = (c > d) ? c : d;
        }
    }
}

// Range-max of key over inclusive element range [a, b] (a <= b).
__device__ __forceinline__ unsigned long long rangeMaxK(
        const float* __restrict__ f,
        const unsigned long long* __restrict__ maxTab,
        int a, int b) {
    unsigned long long best = 0ull;
    int ba = a >> 8, bb = b >> 8;
    if (ba == bb) {
        for (int j = a; j <= b; ++j) {
            unsigned long long k = kkey(f[j], (unsigned int)j);
            if (k > best) best = k;
        }
        return best;
    }
    int e0 = (ba << 8) + (BSZ - 1);
    for (int j = a; j <= e0; ++j) {
        unsigned long long k = kkey(f[j], (unsigned int)j);
        if (k > best) best = k;
    }
    int s1 = bb << 8;
    for (int j = s1; j <= b; ++j) {
        unsigned long long k = kkey(f[j], (unsigned int)j);
        if (k > best) best = k;
    }
    int lo = ba + 1, hi = bb - 1;
    if (lo <= hi) {
        int len = hi - lo + 1;
        int lg = 31 - __clz(len);
        unsigned long long m1 = maxTab[lg * NB + lo];
        unsigned long long m2 = maxTab[lg * NB + hi - (1 << lg) + 1];
        if (m1 > best) best = m1;
        if (m2 > best) best = m2;
    }
    return best;
}

// ---------------- Kernel C: one thread per vertex --------------------------
__global__ void h0_pairs_kernel(const float* __restrict__ f,
                                const unsigned long long* __restrict__ minTab,
                                const unsigned long long* __restrict__ maxTab,
                                float* __restrict__ out) {
    int i = blockIdx.x * blockDim.x + threadIdx.x;
    if (i >= P_N) return;
    float fi = f[i];
    unsigned long long k = kkey(fi, (unsigned int)i);

    // Local-minimum test in stable filtration order (tie -> lower index older).
    bool lm = true;
    if (i > 0)        { if (!(kkey(f[i - 1], (unsigned int)(i - 1)) > k)) lm = false; }
    if (i < P_N - 1)  { if (!(kkey(f[i + 1], (unsigned int)(i + 1)) > k)) lm = false; }

    if (!lm) {  // zero-persistence pair (f[i], f[i])
        reinterpret_cast<float2*>(out)[i] = make_float2(fi, fi);
        return;
    }

    int blk = i >> 8;
    int bstart = blk << 8;

    // ---- nearest index l < i with key < k -------------------------------
    int l = -1;
    for (int j = i - 1; j >= bstart; --j) {
        if (kkey(f[j], (unsigned int)j) < k) { l = j; break; }
    }
    if (l < 0 && blk > 0) {
        int p = blk;  // skip maximal non-qualifying power-of-two block runs
        #pragma unroll
        for (int lg = LEVELS - 1; lg >= 0; --lg) {
            int step = 1 << lg;
            if (p >= step && minTab[lg * NB + (p - step)] >= k) p -= step;
        }
        if (p > 0) {  // block p-1 contains the nearest-smaller; take rightmost
            int ts = (p - 1) << 8;
            __builtin_prefetch(f + ts, 0, 0);            // global_prefetch_b8
            for (int j = ts + BSZ - 1; j >= ts; --j) {
                if (kkey(f[j], (unsigned int)j) < k) { l = j; break; }
            }
        }
    }

    // ---- nearest index r > i with key < k -------------------------------
    int r = -1;
    int bend = bstart + BSZ - 1;
    for (int j = i + 1; j <= bend; ++j) {
        if (kkey(f[j], (unsigned int)j) < k) { r = j; break; }
    }
    if (r < 0 && blk < NB - 1) {
        int q = blk + 1;
        #pragma unroll
        for (int lg = LEVELS - 1; lg >= 0; --lg) {
            int step = 1 << lg;
            if (q + step <= NB && minTab[lg * NB + q] >= k) q += step;
        }
        if (q < NB) {  // block q contains the nearest-smaller; take leftmost
            int ts = q << 8;
            __builtin_prefetch(f + ts, 0, 0);            // global_prefetch_b8
            for (int j = ts; j < ts + BSZ; ++j) {
                if (kkey(f[j], (unsigned int)j) < k) { r = j; break; }
            }
        }
    }

    float death;
    if (l < 0 && r < 0) {
        death = __builtin_inff();  // global minimum -> essential class
    } else {
        unsigned long long kBL = ~0ull, kBR = ~0ull;
        if (l >= 0) kBL = rangeMaxK(f, maxTab, l + 1, i - 1);  // l <= i-2 (local min)
        if (r >= 0) kBR = rangeMaxK(f, maxTab, i + 1, r - 1);  // r >= i+2
        unsigned long long s = (kBL < kBR) ? kBL : kBR;        // elder rule saddle
        unsigned int sidx = (unsigned int)(s & 0xFFFFFFFFull);
        death = f[sidx];
    }
    reinterpret_cast<float2*>(out)[i] = make_float2(fi, death);
}

extern "C" void kernel_launch(void* const* d_in, const int* in_sizes, int n_in,
                              void* d_out, int out_size, void* d_ws, size_t ws_size,
                              hipStream_t stream) {
    (void)in_sizes; (void)n_in; (void)out_size; (void)ws_size;
    const float* f = (const float*)d_in[0];
    float* out = (float*)d_out;
    unsigned long long* minTab = (unsigned long long*)d_ws;               // 14*NB u64
    unsigned long long* maxTab = minTab + (size_t)LEVELS * NB;            // 14*NB u64
    // total ws use: 2 * 14 * 8192 * 8 B = 1.75 MB

    block_minmax_kernel<<<NB, BSZ, 0, stream>>>(f, minTab, maxTab);
    build_sparse_kernel<<<1, 1024, 0, stream>>>(minTab, maxTab);
    h0_pairs_kernel<<<P_N / BSZ, BSZ, 0, stream>>>(f, minTab, maxTab, out);
}